// Prompt_20091857011401
// MI455X (gfx1250) — compile-verified
//
#include <hip/hip_runtime.h>
#include <math.h>

#define B_   256
#define S_   196
#define D_   768
#define P_   1024
#define L_   8
#define K_   8
#define APAD 772   // 16-row A tile padded: (772 % 64)=4 -> conflict-free column reads

typedef float v2f __attribute__((ext_vector_type(2)));
typedef float v4f __attribute__((ext_vector_type(4)));
typedef float v8f __attribute__((ext_vector_type(8)));

// ---------------- 1) mean over S + L2 normalize -> x_norm [B, D] ----------------
__global__ void mean_norm_kernel(const float* __restrict__ x, float* __restrict__ xnorm) {
    int b = blockIdx.x, t = threadIdx.x;
    const float* base = x + (size_t)b * S_ * D_;
    float s0 = 0.f, s1 = 0.f, s2 = 0.f;
    for (int s = 0; s < S_; ++s) {
        const float* r = base + (size_t)s * D_;
        s0 += r[t]; s1 += r[t + 256]; s2 += r[t + 512];
    }
    const float inv = 1.0f / (float)S_;
    s0 *= inv; s1 *= inv; s2 *= inv;
    __shared__ float red[256];
    red[t] = s0 * s0 + s1 * s1 + s2 * s2;
    __syncthreads();
    for (int st = 128; st > 0; st >>= 1) { if (t < st) red[t] += red[t + st]; __syncthreads(); }
    float sc = rsqrtf(fmaxf(red[0], 1e-12f));
    float* o = xnorm + (size_t)b * D_;
    o[t] = s0 * sc; o[t + 256] = s1 * sc; o[t + 512] = s2 * sc;
}

// ---------------- 2) L2 normalize prompt_key -> key_norm [P, D] ----------------
__global__ void key_norm_kernel(const float* __restrict__ pk, float* __restrict__ knorm) {
    int p = blockIdx.x, t = threadIdx.x;
    const float* r = pk + (size_t)p * D_;
    float v0 = r[t], v1 = r[t + 256], v2 = r[t + 512];
    __shared__ float red[256];
    red[t] = v0 * v0 + v1 * v1 + v2 * v2;
    __syncthreads();
    for (int st = 128; st > 0; st >>= 1) { if (t < st) red[t] += red[t + st]; __syncthreads(); }
    float sc = rsqrtf(fmaxf(red[0], 1e-12f));
    float* o = knorm + (size_t)p * D_;
    o[t] = v0 * sc; o[t + 256] = v1 * sc; o[t + 512] = v2 * sc;
}

// ---------------- 3) similarity = x_norm @ key_norm^T via V_WMMA_F32_16X16X4_F32 ----------------
// grid.x = B/16 (M tiles), grid.y = P/128; 8 waves/block, each wave owns one 16x16 output tile.
__global__ void sim_wmma_kernel(const float* __restrict__ xnorm, const float* __restrict__ knorm,
                                float* __restrict__ sim) {
    __shared__ float As[16][APAD];
    const int tid = threadIdx.x;
    const int m0  = blockIdx.x * 16;
    for (int i = tid; i < 16 * D_; i += 256) {
        int r = i / D_, c = i - r * D_;
        As[r][c] = xnorm[(size_t)(m0 + r) * D_ + c];
    }
    __syncthreads();

    const int wave = tid >> 5, lane = tid & 31;
    const int n0   = blockIdx.y * 128 + wave * 16;
    const int arow = lane & 15;            // A: M = lane&15
    const int koff = (lane >> 4) * 2;      // lanes 0-15 -> K=k,k+1 ; lanes 16-31 -> K=k+2,k+3
    const int bcol = n0 + (lane & 15);     // B: N = n0 + lane&15
    const float* bbase = knorm + (size_t)bcol * D_ + koff;

    v8f c = {};
    for (int k = 0; k < D_; k += 4) {
        v2f a;
        a.x = As[arow][k + koff];
        a.y = As[arow][k + koff + 1];
        v2f bv = *(const v2f*)(bbase + k);   // 8B aligned (k+koff even)
        c = __builtin_amdgcn_wmma_f32_16x16x4_f32(false, a, false, bv, (short)0, c, false, false);
    }
    const int col   = n0 + (lane & 15);
    const int rbase = m0 + ((lane >> 4) << 3);   // C: M = v + 8*(lane>=16), N = lane&15
#pragma unroll
    for (int v = 0; v < 8; ++v)
        sim[(size_t)(rbase + v) * P_ + col] = c[v];
}

// ---------------- 4) per-row top-8 (value desc, index asc on ties) ----------------
__global__ void topk_rows_kernel(const float* __restrict__ sim, int* __restrict__ idx_rows) {
    const int b = blockIdx.x, t = threadIdx.x;
    float v[4]; int ci[4];
#pragma unroll
    for (int j = 0; j < 4; ++j) { ci[j] = t + j * 256; v[j] = sim[(size_t)b * P_ + ci[j]]; }
    __shared__ float bval[256];
    __shared__ int   bidx[256];
    for (int it = 0; it < K_; ++it) {
        float lv = v[0]; int li = ci[0];
#pragma unroll
        for (int j = 1; j < 4; ++j)
            if (v[j] > lv || (v[j] == lv && ci[j] < li)) { lv = v[j]; li = ci[j]; }
        bval[t] = lv; bidx[t] = li;
        __syncthreads();
        for (int st = 128; st > 0; st >>= 1) {
            if (t < st) {
                if (bval[t + st] > bval[t] ||
                    (bval[t + st] == bval[t] && bidx[t + st] < bidx[t])) {
                    bval[t] = bval[t + st]; bidx[t] = bidx[t + st];
                }
            }
            __syncthreads();
        }
        int best = bidx[0];
        if (t == 0) idx_rows[b * K_ + it] = best;
        if (t == (best & 255)) v[best >> 8] = -INFINITY;   // owner masks the winner
        __syncthreads();                                    // protect bval/bidx reuse
    }
}

// ---------------- 5) bincount + consensus top-8 + broadcast idx output ----------------
__global__ void consensus_kernel(const int* __restrict__ idx_rows, int* __restrict__ major,
                                 float* __restrict__ idx_out) {
    __shared__ int       cnt[P_];
    __shared__ long long key[P_];
    __shared__ int       majL[K_];
    const int t = threadIdx.x;                 // 1024 threads
    cnt[t] = 0;
    __syncthreads();
    atomicAdd(&cnt[idx_rows[t]], 1);           // 2048 entries, 2 per thread
    atomicAdd(&cnt[idx_rows[t + 1024]], 1);
    __syncthreads();
    for (int k = 0; k < K_; ++k) {
        key[t] = (((long long)cnt[t]) << 16) | (long long)(1023 - t);  // count desc, idx asc
        __syncthreads();
        for (int st = 512; st > 0; st >>= 1) {
            if (t < st && key[t + st] > key[t]) key[t] = key[t + st];
            __syncthreads();
        }
        long long best = key[0];
        __syncthreads();
        int bi = 1023 - (int)(best & 0xFFFF);
        if (t == 0) { major[k] = bi; majL[k] = bi; }
        if (t == bi) cnt[t] = -1;              // negative key -> never wins again
        __syncthreads();
    }
    for (int i = t; i < B_ * K_; i += 1024)    // idx output = broadcast major ids, as float
        idx_out[i] = (float)majL[i & 7];
}

// ---------------- 6) reduce_sim = (1/B) * sum_b sum_k similarity[b, major_k] ----------------
__global__ void reduce_sim_kernel(const float* __restrict__ sim, const int* __restrict__ major,
                                  float* __restrict__ out_scalar) {
    const int t = threadIdx.x;                 // t == batch index
    float s = 0.f;
#pragma unroll
    for (int k = 0; k < K_; ++k) s += sim[(size_t)t * P_ + major[k]];
    __shared__ float red[256];
    red[t] = s;
    __syncthreads();
    for (int st = 128; st > 0; st >>= 1) { if (t < st) red[t] += red[t + st]; __syncthreads(); }
    if (t == 0) *out_scalar = red[0] / (float)B_;
}

// ---------------- 7) concat [prompt[major] (64 rows) | x_embed (196 rows)] per batch ----------------
__global__ void assemble_kernel(const float* __restrict__ x, const float* __restrict__ prompt,
                                const int* __restrict__ major, float* __restrict__ out) {
    const int D4  = D_ / 4;                    // 192 x 16B per row
    const int i   = blockIdx.x * 256 + threadIdx.x;   // v4f index, exact coverage
    const int row = i / D4;
    const int d4  = i - row * D4;
    const int b   = row / (K_ * L_ + S_);
    const int r   = row - b * (K_ * L_ + S_);
    v4f val;
    if (r < K_ * L_) {
        int pid = major[r >> 3];
        val = ((const v4f*)prompt)[((size_t)pid * L_ + (r & 7)) * D4 + d4];
    } else {
        val = ((const v4f*)x)[((size_t)b * S_ + (r - K_ * L_)) * D4 + d4];
    }
    __builtin_nontemporal_store(val, (v4f*)out + i);   // 205MB streamed, never re-read
}

extern "C" void kernel_launch(void* const* d_in, const int* in_sizes, int n_in,
                              void* d_out, int out_size, void* d_ws, size_t ws_size,
                              hipStream_t stream) {
    const float* x      = (const float*)d_in[0];   // [B, S, D]
    const float* prompt = (const float*)d_in[1];   // [P, L, D]
    const float* pkey   = (const float*)d_in[2];   // [P, D]

    float* out      = (float*)d_out;
    float* out_emb  = out;                                        // [B, 260, D]
    float* out_rsim = out + (size_t)B_ * (K_ * L_ + S_) * D_;     // scalar
    float* out_sim  = out_rsim + 1;                               // [B, P]
    float* out_idx  = out_sim + (size_t)B_ * P_;                  // [B, K] as float

    float* xnorm    = (float*)d_ws;                // B*D   = 196608 floats
    float* knorm    = xnorm + (size_t)B_ * D_;     // P*D   = 786432 floats
    int*   idx_rows = (int*)(knorm + (size_t)P_ * D_);  // B*K = 2048 ints
    int*   major    = idx_rows + B_ * K_;          // 8 ints

    mean_norm_kernel <<<B_,               256, 0, stream>>>(x, xnorm);
    key_norm_kernel  <<<P_,               256, 0, stream>>>(pkey, knorm);
    sim_wmma_kernel  <<<dim3(B_/16, P_/128), 256, 0, stream>>>(xnorm, knorm, out_sim);
    topk_rows_kernel <<<B_,               256, 0, stream>>>(out_sim, idx_rows);
    consensus_kernel <<<1,               1024, 0, stream>>>(idx_rows, major, out_idx);
    reduce_sim_kernel<<<1,                256, 0, stream>>>(out_sim, major, out_rsim);
    assemble_kernel  <<<(B_*(K_*L_+S_)*D_/4)/256, 256, 0, stream>>>(x, prompt, major, out_emb);
}